// JointNetwork_62775241999026
// MI455X (gfx1250) — compile-verified
//
#include <hip/hip_runtime.h>
#include <math.h>

typedef float v2f __attribute__((ext_vector_type(2)));
typedef float v8f __attribute__((ext_vector_type(8)));

#define HH 256
#define VV 1024
#define BB 8
#define TT 256
#define UU 64

// ---------------------------------------------------------------------------
// Phase 1: C[M, V] = A[M, H] * W[H, V]   (fp32, exact, via V_WMMA_F32_16X16X4_F32)
// One wave computes a 16x64 output tile: 4 accumulators, A-fragment reused 4x.
// ---------------------------------------------------------------------------
__global__ void __launch_bounds__(256)
proj_gemm_kernel(const float* __restrict__ A, const float* __restrict__ W,
                 float* __restrict__ C, int Mtiles) {
  const int wave  = (blockIdx.x * blockDim.x + threadIdx.x) >> 5;
  const int lane  = threadIdx.x & 31;
  const int tileM = wave >> 4;       // 16 column-groups of 64 along V=1024
  const int grp   = wave & 15;
  if (tileM >= Mtiles) return;       // wave-uniform: EXEC stays all-ones

  const int half = lane >> 4;        // 0: lanes 0-15, 1: lanes 16-31
  const int l15  = lane & 15;

  const float* Arow   = A + (tileM * 16 + l15) * HH;   // A fragment rows
  const int   colBase = grp * 64 + l15;                // B/D columns

  v8f c0 = {}, c1 = {}, c2 = {}, c3 = {};
  #pragma unroll 4
  for (int k = 0; k < HH; k += 4) {
    // A 16x4 fragment: v0 = K=k (+2*half), v1 = K=k+1 (+2*half)
    v2f a;
    a.x = Arow[k + 2 * half];
    a.y = Arow[k + 2 * half + 1];
    // B 4x16 fragments (row k striped across lanes within a VGPR)
    const float* Wr0 = W + (size_t)(k + 2 * half) * VV;
    const float* Wr1 = Wr0 + VV;
    v2f b0, b1, b2, b3;
    b0.x = Wr0[colBase +  0]; b0.y = Wr1[colBase +  0];
    b1.x = Wr0[colBase + 16]; b1.y = Wr1[colBase + 16];
    b2.x = Wr0[colBase + 32]; b2.y = Wr1[colBase + 32];
    b3.x = Wr0[colBase + 48]; b3.y = Wr1[colBase + 48];
    c0 = __builtin_amdgcn_wmma_f32_16x16x4_f32(false, a, false, b0, (short)0, c0, false, false);
    c1 = __builtin_amdgcn_wmma_f32_16x16x4_f32(false, a, false, b1, (short)0, c1, false, false);
    c2 = __builtin_amdgcn_wmma_f32_16x16x4_f32(false, a, false, b2, (short)0, c2, false, false);
    c3 = __builtin_amdgcn_wmma_f32_16x16x4_f32(false, a, false, b3, (short)0, c3, false, false);
  }

  // D 16x16 f32 layout: VGPR r -> M=r (lanes 0-15) / M=r+8 (lanes 16-31)
  #pragma unroll
  for (int r = 0; r < 8; ++r) {
    const int m = tileM * 16 + r + half * 8;
    float* Cr = C + (size_t)m * VV;
    Cr[colBase +  0] = c0[r];
    Cr[colBase + 16] = c1[r];
    Cr[colBase + 32] = c2[r];
    Cr[colBase + 48] = c3[r];
  }
}

// ---------------------------------------------------------------------------
// Phase 2: out[b,t,u,:] = log_softmax(encP[b,t,:] + decP[b,u,:] + bias)
// One wave per row of V=1024: 32 fp32/lane in registers (single pass over x),
// shfl_xor butterfly reductions, non-temporal streaming stores (512 MiB).
// encP/decP/bias stay resident in the 192 MB L2.
// ---------------------------------------------------------------------------
__global__ void __launch_bounds__(256)
joint_logsoftmax_kernel(const float* __restrict__ encP, const float* __restrict__ decP,
                        const float* __restrict__ bias, float* __restrict__ out) {
  const int row  = (int)((blockIdx.x * blockDim.x + threadIdx.x) >> 5); // 131072 rows
  const int lane = threadIdx.x & 31;

  const int u = row % UU;
  const int t = (row / UU) % TT;
  const int b = row / (UU * TT);

  const float* er = encP + (size_t)(b * TT + t) * VV;
  const float* dr = decP + (size_t)(b * UU + u) * VV;

  float x[32];
  float mx = -INFINITY;
  #pragma unroll
  for (int i = 0; i < 32; ++i) {
    const int v = lane + (i << 5);              // stride-32: coalesced
    const float val = er[v] + dr[v] + bias[v];
    x[i] = val;
    mx = fmaxf(mx, val);
  }
  #pragma unroll
  for (int off = 16; off > 0; off >>= 1)
    mx = fmaxf(mx, __shfl_xor(mx, off, 32));

  float s = 0.0f;
  #pragma unroll
  for (int i = 0; i < 32; ++i)
    s += __expf(x[i] - mx);
  #pragma unroll
  for (int off = 16; off > 0; off >>= 1)
    s += __shfl_xor(s, off, 32);

  const float lse = mx + __logf(s);

  float* o = out + (size_t)row * VV;
  #pragma unroll
  for (int i = 0; i < 32; ++i)
    __builtin_nontemporal_store(x[i] - lse, o + lane + (i << 5));
}

// ---------------------------------------------------------------------------
extern "C" void kernel_launch(void* const* d_in, const int* in_sizes, int n_in,
                              void* d_out, int out_size, void* d_ws, size_t ws_size,
                              hipStream_t stream) {
  const float* enc  = (const float*)d_in[0];   // [8,256,256]
  const float* dec  = (const float*)d_in[1];   // [8,64,256]
  const float* W    = (const float*)d_in[2];   // [512,1024]
  const float* bias = (const float*)d_in[3];   // [1024]
  float* out = (float*)d_out;                  // [8,256,64,1024]

  float* encP = (float*)d_ws;                          // 2048*1024 fp32 = 8 MiB
  float* decP = encP + (size_t)BB * TT * VV;           //  512*1024 fp32 = 2 MiB

  // enc_proj: M=2048 -> 128 M-tiles * 16 col-groups = 2048 waves = 256 blocks
  proj_gemm_kernel<<<256, 256, 0, stream>>>(enc, W, encP, BB * TT / 16);
  // dec_proj: M=512 -> 32 M-tiles * 16 col-groups = 512 waves = 64 blocks
  proj_gemm_kernel<<<64, 256, 0, stream>>>(dec, W + (size_t)HH * VV, decP, BB * UU / 16);

  // 131072 rows, 8 waves/block -> 16384 blocks
  joint_logsoftmax_kernel<<<16384, 256, 0, stream>>>(encP, decP, bias, out);
}